// Simple_Model_36155034698522
// MI455X (gfx1250) — compile-verified
//
#include <hip/hip_runtime.h>
#include <hip/hip_bf16.h>
#include <math.h>

typedef float v2f __attribute__((ext_vector_type(2)));
typedef float v8f __attribute__((ext_vector_type(8)));

#define NLAYERS 4
#define DCH 24

// ---------------------------------------------------------------------------
// init kernels
// ---------------------------------------------------------------------------
__global__ void k_layer_init(float* __restrict__ agg, float* __restrict__ maxv,
                             float* __restrict__ denom, int n) {
  int i = blockIdx.x * blockDim.x + threadIdx.x;
  if (i < n * DCH) agg[i] = 0.0f;
  if (i < n) { maxv[i] = -INFINITY; denom[i] = 0.0f; }
}

__global__ void k_zero(float* __restrict__ p, int n) {
  int i = blockIdx.x * blockDim.x + threadIdx.x;
  if (i < n) p[i] = 0.0f;
}

// ---------------------------------------------------------------------------
// WMMA GEMM: out[N,24] = H[N,24] @ W[24,24] + bias   (f32, K split into 6x4)
// block = 256 (8 waves); wave w -> node tile blockIdx.x*8+w
// grid.y = 2 column tiles; grid.z = 4 weight matrices (q,k,v,skip)
// A layout (16x4 f32): lane<16 -> row=lane, K={0,1}; lane>=16 -> row=lane-16, K={2,3}
// B layout (4x16 f32): lane<16 -> col=lane, K={0,1}; lane>=16 -> col=lane-16, K={2,3}
// C/D layout: vgpr i, lane L -> row = i + (L>=16)*8, col = L%16
// Out-of-range rows/cols load clamped addresses (harmless: their WMMA output
// rows/cols are never stored), keeping every load unconditional.
// ---------------------------------------------------------------------------
struct GemmArgs {
  const float* W[4];
  const float* bias[4];
  float* out[4];
};

__global__ void k_gemm24(const float* __restrict__ H, GemmArgs args, int n) {
  const int wave  = threadIdx.x >> 5;
  const int lane  = threadIdx.x & 31;
  const int tile  = blockIdx.x * 8 + wave;
  if (tile * 16 >= n) return;               // wave-uniform exit

  const int mat   = blockIdx.z;
  const int ntile = blockIdx.y;
  const float* __restrict__ W    = args.W[mat];
  const float* __restrict__ bias = args.bias[mat];
  float* __restrict__ out        = args.out[mat];

  const int m    = lane & 15;
  const int half = lane >> 4;               // 0 or 1
  const int koff = half * 2;                // K sub-offset within a 4-step
  const int col  = ntile * 16 + m;          // output / B column
  const int colC = col < DCH ? col : (DCH - 1);
  const int node = tile * 16 + m;
  const int nodeC = node < n ? node : (n - 1);

  // hoisted, unconditional fragment loads
  v2f a[6], b[6];
#pragma unroll
  for (int ks = 0; ks < 6; ++ks) {
    const int k0 = ks * 4 + koff;           // even -> 8B aligned
    a[ks] = *(const v2f*)(H + (size_t)nodeC * DCH + k0);
    b[ks].x = W[k0 * DCH + colC];
    b[ks].y = W[(k0 + 1) * DCH + colC];
  }

  v8f c = {};
#pragma unroll
  for (int ks = 0; ks < 6; ++ks)
    c = __builtin_amdgcn_wmma_f32_16x16x4_f32(false, a[ks], false, b[ks],
                                              (short)0, c, false, false);

  if (col < DCH) {
    const float bb = bias[col];
    if (tile * 16 + 16 <= n) {              // fast path: full tile
#pragma unroll
      for (int i = 0; i < 8; ++i)
        out[(size_t)(tile * 16 + i + half * 8) * DCH + col] = c[i] + bb;
    } else {
#pragma unroll
      for (int i = 0; i < 8; ++i) {
        const int row = tile * 16 + i + half * 8;
        if (row < n) out[(size_t)row * DCH + col] = c[i] + bb;
      }
    }
  }
}

// ---------------------------------------------------------------------------
// edge kernels
// ---------------------------------------------------------------------------
__device__ inline void atomicMaxFloat(float* addr, float val) {
  // positive floats ordered as ints; negative floats reverse-ordered as uints.
  if (val >= 0.0f) atomicMax((int*)addr, __float_as_int(val));
  else             atomicMin((unsigned int*)addr, __float_as_uint(val));
}

__global__ void k_edge_logits(const float* __restrict__ q, const float* __restrict__ k,
                              const int* __restrict__ src, const int* __restrict__ dst,
                              float* __restrict__ elog, float* __restrict__ maxv, int nE) {
  int e = blockIdx.x * blockDim.x + threadIdx.x;
  if (e >= nE) return;
  const int s = src[e], d = dst[e];
  const float4* __restrict__ qv = (const float4*)(q + (size_t)d * DCH);
  const float4* __restrict__ kv = (const float4*)(k + (size_t)s * DCH);
  float acc = 0.0f;
#pragma unroll
  for (int j = 0; j < 6; ++j) {
    float4 qa = qv[j], ka = kv[j];
    acc += qa.x * ka.x + qa.y * ka.y + qa.z * ka.z + qa.w * ka.w;
  }
  acc *= 0.2041241452319315f;  // 1/sqrt(24)
  elog[e] = acc;
  atomicMaxFloat(&maxv[d], acc);
}

__global__ void k_edge_expsum(float* __restrict__ eval, const float* __restrict__ maxv,
                              const int* __restrict__ dst, float* __restrict__ denom, int nE) {
  int e = blockIdx.x * blockDim.x + threadIdx.x;
  if (e >= nE) return;
  const int d = dst[e];
  const float ev = expf(eval[e] - maxv[d]);
  eval[e] = ev;
  atomicAdd(&denom[d], ev);
}

__global__ void k_edge_scatter(const float* __restrict__ eval, const float* __restrict__ denom,
                               const float* __restrict__ v, const int* __restrict__ src,
                               const int* __restrict__ dst, float* __restrict__ agg, int nE) {
  int e = blockIdx.x * blockDim.x + threadIdx.x;
  if (e >= nE) return;
  const int s = src[e], d = dst[e];
  const float coef = eval[e] / (denom[d] + 1e-16f);
  const float4* __restrict__ vv = (const float4*)(v + (size_t)s * DCH);
  float* __restrict__ ag = agg + (size_t)d * DCH;
#pragma unroll
  for (int j = 0; j < 6; ++j) {
    float4 va = vv[j];
    atomicAdd(&ag[j * 4 + 0], coef * va.x);
    atomicAdd(&ag[j * 4 + 1], coef * va.y);
    atomicAdd(&ag[j * 4 + 2], coef * va.z);
    atomicAdd(&ag[j * 4 + 3], coef * va.w);
  }
}

__global__ void k_node_update(const float* __restrict__ agg, const float* __restrict__ skip,
                              float* __restrict__ hout, int n) {
  int i = blockIdx.x * blockDim.x + threadIdx.x;
  if (i >= n) return;
  float t = agg[i] + skip[i];
  hout[i] = t > 0.0f ? t : 0.0f;
}

// ---------------------------------------------------------------------------
// pooling + head
// ---------------------------------------------------------------------------
__global__ void k_pool(const float* __restrict__ h, const int* __restrict__ batch,
                       float* __restrict__ sums, float* __restrict__ counts, int n) {
  int i = blockIdx.x * blockDim.x + threadIdx.x;
  if (i >= n) return;
  const int b = batch[i];
  const float4* __restrict__ hv = (const float4*)(h + (size_t)i * DCH);
  float* __restrict__ sm = sums + (size_t)b * DCH;
#pragma unroll
  for (int j = 0; j < 6; ++j) {
    float4 ha = hv[j];
    atomicAdd(&sm[j * 4 + 0], ha.x);
    atomicAdd(&sm[j * 4 + 1], ha.y);
    atomicAdd(&sm[j * 4 + 2], ha.z);
    atomicAdd(&sm[j * 4 + 3], ha.w);
  }
  atomicAdd(&counts[b], 1.0f);
}

__global__ void k_head(const float* __restrict__ sums, const float* __restrict__ counts,
                       const float* __restrict__ gf,
                       const float* __restrict__ gfW1, const float* __restrict__ gfb1,
                       const float* __restrict__ gfW2, const float* __restrict__ gfb2,
                       const float* __restrict__ W1, const float* __restrict__ b1,
                       const float* __restrict__ W2, const float* __restrict__ b2,
                       const float* __restrict__ W3, const float* __restrict__ b3,
                       float* __restrict__ out) {
  __shared__ float z[36];
  __shared__ float g1[12];
  __shared__ float h1[128];
  __shared__ float h2[128];
  __shared__ float red[128];
  const int g = blockIdx.x;
  const int t = threadIdx.x;

  if (t < 24) {
    float cnt = counts[g];
    cnt = cnt > 1.0f ? cnt : 1.0f;
    z[t] = sums[g * DCH + t] / cnt;
  }
  if (t < 12) {
    float acc = gfb1[t];
#pragma unroll
    for (int i = 0; i < 6; ++i) acc += gf[g * 6 + i] * gfW1[i * 12 + t];
    g1[t] = acc > 0.0f ? acc : 0.0f;
  }
  __syncthreads();
  if (t < 12) {
    float acc = gfb2[t];
#pragma unroll
    for (int i = 0; i < 12; ++i) acc += g1[i] * gfW2[i * 12 + t];
    z[24 + t] = acc > 0.0f ? acc : 0.0f;
  }
  __syncthreads();
  {
    float acc = b1[t];
#pragma unroll
    for (int i = 0; i < 36; ++i) acc += z[i] * W1[i * 128 + t];
    h1[t] = acc > 0.0f ? acc : 0.0f;
  }
  __syncthreads();
  {
    float acc = b2[t];
    for (int i = 0; i < 128; ++i) acc += h1[i] * W2[i * 128 + t];
    h2[t] = acc > 0.0f ? acc : 0.0f;
  }
  __syncthreads();
  red[t] = h2[t] * W3[t];
  __syncthreads();
  for (int s = 64; s > 0; s >>= 1) {
    if (t < s) red[t] += red[t + s];
    __syncthreads();
  }
  if (t == 0) out[g] = red[0] + b3[0];
}

// ---------------------------------------------------------------------------
// launch
// ---------------------------------------------------------------------------
extern "C" void kernel_launch(void* const* d_in, const int* in_sizes, int n_in,
                              void* d_out, int out_size, void* d_ws, size_t ws_size,
                              hipStream_t stream) {
  const float* x   = (const float*)d_in[0];
  const int*   ei  = (const int*)d_in[1];
  const int*   bat = (const int*)d_in[2];
  const float* gf  = (const float*)d_in[3];
  const float* Wq  = (const float*)d_in[4];
  const float* bq  = (const float*)d_in[5];
  const float* Wk  = (const float*)d_in[6];
  const float* bk  = (const float*)d_in[7];
  const float* Wv  = (const float*)d_in[8];
  const float* bv  = (const float*)d_in[9];
  const float* Ws  = (const float*)d_in[10];
  const float* bs  = (const float*)d_in[11];
  const float* gfW1 = (const float*)d_in[12];
  const float* gfb1 = (const float*)d_in[13];
  const float* gfW2 = (const float*)d_in[14];
  const float* gfb2 = (const float*)d_in[15];
  const float* W1  = (const float*)d_in[16];
  const float* b1  = (const float*)d_in[17];
  const float* W2  = (const float*)d_in[18];
  const float* b2  = (const float*)d_in[19];
  const float* W3  = (const float*)d_in[20];
  const float* b3  = (const float*)d_in[21];

  const int N = in_sizes[0] / DCH;       // 100000
  const int E = in_sizes[1] / 2;         // 3200000
  const int G = in_sizes[3] / 6;         // 512
  const int* src = ei;
  const int* dst = ei + E;

  // workspace layout (floats)
  float* ws = (float*)d_ws;
  size_t off = 0;
  float* h0    = ws + off; off += (size_t)N * DCH;
  float* h1b   = ws + off; off += (size_t)N * DCH;
  float* qb    = ws + off; off += (size_t)N * DCH;
  float* kb    = ws + off; off += (size_t)N * DCH;
  float* vb    = ws + off; off += (size_t)N * DCH;
  float* sb    = ws + off; off += (size_t)N * DCH;
  float* agg   = ws + off; off += (size_t)N * DCH;
  float* maxv  = ws + off; off += (size_t)N;
  float* denom = ws + off; off += (size_t)N;
  float* eval  = ws + off; off += (size_t)E;
  float* sums  = ws + off; off += (size_t)G * DCH;
  float* cnts  = ws + off; off += (size_t)G;

  const int TB = 256;
  dim3 gemmGrid((N + 16 * 8 - 1) / (16 * 8), 2, 4);

  const float* hcur = x;
  float* hbuf[2] = {h0, h1b};

  for (int l = 0; l < NLAYERS; ++l) {
    k_layer_init<<<(N * DCH + TB - 1) / TB, TB, 0, stream>>>(agg, maxv, denom, N);

    GemmArgs ga;
    ga.W[0] = Wq + l * DCH * DCH; ga.bias[0] = bq + l * DCH; ga.out[0] = qb;
    ga.W[1] = Wk + l * DCH * DCH; ga.bias[1] = bk + l * DCH; ga.out[1] = kb;
    ga.W[2] = Wv + l * DCH * DCH; ga.bias[2] = bv + l * DCH; ga.out[2] = vb;
    ga.W[3] = Ws + l * DCH * DCH; ga.bias[3] = bs + l * DCH; ga.out[3] = sb;
    k_gemm24<<<gemmGrid, 256, 0, stream>>>(hcur, ga, N);

    k_edge_logits<<<(E + TB - 1) / TB, TB, 0, stream>>>(qb, kb, src, dst, eval, maxv, E);
    k_edge_expsum<<<(E + TB - 1) / TB, TB, 0, stream>>>(eval, maxv, dst, denom, E);
    k_edge_scatter<<<(E + TB - 1) / TB, TB, 0, stream>>>(eval, denom, vb, src, dst, agg, E);

    k_node_update<<<(N * DCH + TB - 1) / TB, TB, 0, stream>>>(agg, sb, hbuf[l & 1], N * DCH);
    hcur = hbuf[l & 1];
  }

  k_zero<<<(G * DCH + G + TB - 1) / TB, TB, 0, stream>>>(sums, G * DCH + G); // sums then cnts (contiguous)
  k_pool<<<(N + TB - 1) / TB, TB, 0, stream>>>(hcur, bat, sums, cnts, N);
  k_head<<<G, 128, 0, stream>>>(sums, cnts, gf, gfW1, gfb1, gfW2, gfb2,
                                W1, b1, W2, b2, W3, b3, (float*)d_out);
}